// E25CellAutograd_2147483648037
// MI455X (gfx1250) — compile-verified
//
#include <hip/hip_runtime.h>
#include <math.h>

// ---------------- problem constants ----------------
#define DIM     512
#define NSLOT   128
#define BATCH   256
#define TSTEPS  256
#define TS      513            // padded LDS tape row stride (floats): odd # banks

typedef __attribute__((ext_vector_type(2))) float v2f;
typedef __attribute__((ext_vector_type(8))) float v8f;

// score scale: (1/sqrt(512)) * 0.5  (entmax divides z by 2 up front)
#define ZSCALE 0.02209708691207961f

// =====================================================================
// Kernel 1: x_pre[b,t,:] = x_seq[b,t,:] @ W_x^T + b_h  via V_WMMA_F32_16X16X4_F32
//   Each wave computes one 16-row strip x 4 consecutive 16-col tiles:
//   per K-step -> 1 A-fragment load feeds 4 WMMAs (4x less X traffic).
// =====================================================================
#define NT 4   // n-tiles per wave
__global__ __launch_bounds__(256) void xw_gemm_wmma4(
    const float* __restrict__ X,      // [B*T, DIM]
    const float* __restrict__ W,      // [DIM, DIM]
    const float* __restrict__ bias,   // [DIM]
    float* __restrict__ Y)            // [B*T, DIM]
{
  const int wave  = blockIdx.x * (256 / 32) + (threadIdx.x >> 5);
  const int lane  = threadIdx.x & 31;
  const int ngrp  = wave & 7;             // 8 groups of NT=4 col tiles
  const int mtile = wave >> 3;            // 4096 row tiles
  const int m0 = mtile << 4;
  const int n0 = ngrp * (NT * 16);
  const int half = lane >> 4;             // 0: K pair {0,1}, 1: K pair {2,3}
  const int lm   = lane & 15;             // A row index / B col index
  const int kb   = half << 1;

  const float* xrow  = X + (size_t)(m0 + lm) * DIM;
  const float* wrow0 = W + (size_t)(n0 + 0 * 16 + lm) * DIM;
  const float* wrow1 = W + (size_t)(n0 + 1 * 16 + lm) * DIM;
  const float* wrow2 = W + (size_t)(n0 + 2 * 16 + lm) * DIM;
  const float* wrow3 = W + (size_t)(n0 + 3 * 16 + lm) * DIM;

  v8f c0, c1, c2, c3;
  {
    const float b0 = bias[n0 + 0 * 16 + lm];
    const float b1 = bias[n0 + 1 * 16 + lm];
    const float b2 = bias[n0 + 2 * 16 + lm];
    const float b3 = bias[n0 + 3 * 16 + lm];
#pragma unroll
    for (int i = 0; i < 8; ++i) { c0[i] = b0; c1[i] = b1; c2[i] = b2; c3[i] = b3; }
  }

#pragma unroll 4
  for (int k0 = 0; k0 < DIM; k0 += 4) {
    const v2f a  = *(const v2f*)(xrow  + k0 + kb);
    const v2f b0 = *(const v2f*)(wrow0 + k0 + kb);
    const v2f b1 = *(const v2f*)(wrow1 + k0 + kb);
    const v2f b2 = *(const v2f*)(wrow2 + k0 + kb);
    const v2f b3 = *(const v2f*)(wrow3 + k0 + kb);
    c0 = __builtin_amdgcn_wmma_f32_16x16x4_f32(false, a, false, b0, (short)0, c0, false, false);
    c1 = __builtin_amdgcn_wmma_f32_16x16x4_f32(false, a, false, b1, (short)0, c1, false, false);
    c2 = __builtin_amdgcn_wmma_f32_16x16x4_f32(false, a, false, b2, (short)0, c2, false, false);
    c3 = __builtin_amdgcn_wmma_f32_16x16x4_f32(false, a, false, b3, (short)0, c3, false, false);
  }

  // D layout: vgpr i -> row m0 + i + 8*half, col n0 + j*16 + lm
  float* ybase = Y + (size_t)m0 * DIM + lm;
#pragma unroll
  for (int i = 0; i < 8; ++i) {
    float* yr = ybase + (size_t)(i + 8 * half) * DIM;
    yr[n0 + 0 * 16] = c0[i];
    yr[n0 + 1 * 16] = c1[i];
    yr[n0 + 2 * 16] = c2[i];
    yr[n0 + 3 * 16] = c3[i];
  }
}

// =====================================================================
// Block-cooperative exact 1.5-entmax over 128 values (matches reference).
// z[0..127] holds score*scale*0.5 on entry; attn[0..127] = p*p on exit.
// Must be called by all 512 threads (contains barriers).
// =====================================================================
__device__ __forceinline__ void block_entmax15(
    int tid, float* z, float* zs, float* tau, float* c1, float* c2,
    int* flg, float* attn, float* scal)
{
  __syncthreads();
  // ---- max reduction ----
  if (tid < NSLOT) zs[tid] = z[tid];
  for (int s = NSLOT / 2; s > 0; s >>= 1) {
    __syncthreads();
    if (tid < s) zs[tid] = fmaxf(zs[tid], zs[tid + s]);
  }
  __syncthreads();
  const float mx = zs[0];
  __syncthreads();
  if (tid < NSLOT) { z[tid] -= mx; zs[tid] = z[tid]; }

  // ---- bitonic sort, descending ----
  for (int k = 2; k <= NSLOT; k <<= 1) {
    for (int j = k >> 1; j > 0; j >>= 1) {
      __syncthreads();
      if (tid < NSLOT) {
        const int ixj = tid ^ j;
        if (ixj > tid) {
          const float a = zs[tid], b = zs[ixj];
          const bool up = ((tid & k) == 0);
          if (up ? (a < b) : (a > b)) { zs[tid] = b; zs[ixj] = a; }
        }
      }
    }
  }
  __syncthreads();

  // ---- inclusive scans of zs and zs^2 (Hillis-Steele) ----
  if (tid < NSLOT) { const float v = zs[tid]; c1[tid] = v; c2[tid] = v * v; }
  for (int off = 1; off < NSLOT; off <<= 1) {
    __syncthreads();
    float a = 0.f, b = 0.f;
    if (tid < NSLOT && tid >= off) { a = c1[tid - off]; b = c2[tid - off]; }
    __syncthreads();
    if (tid < NSLOT) { c1[tid] += a; c2[tid] += b; }
  }
  __syncthreads();

  // ---- tau + support flags ----
  if (tid < NSLOT) {
    const float kk   = (float)(tid + 1);
    const float mean = c1[tid] / kk;
    const float msq  = c2[tid] / kk;
    const float ss   = kk * (msq - mean * mean);
    const float dl   = (1.0f - ss) / kk;
    const float tt   = mean - sqrtf(fmaxf(dl, 0.0f));
    tau[tid] = tt;
    flg[tid] = (tt <= zs[tid]) ? 1 : 0;
  }
  for (int s = NSLOT / 2; s > 0; s >>= 1) {
    __syncthreads();
    if (tid < s) flg[tid] += flg[tid + s];
  }
  __syncthreads();
  if (tid == 0) scal[0] = tau[flg[0] - 1];   // tau_star
  __syncthreads();
  if (tid < NSLOT) {
    const float p = fmaxf(z[tid] - scal[0], 0.0f);
    attn[tid] = p * p;
  }
  __syncthreads();
}

// ---- async global->LDS helpers (gfx1250, ASYNCcnt-tracked) ----
__device__ __forceinline__ void async_load_f32_to_lds(const float* gaddr, float* lds) {
  const unsigned lds_off = (unsigned)(size_t)lds;                 // addr[31:0] = LDS byte addr
  const unsigned long long ga = (unsigned long long)(size_t)gaddr;
  asm volatile("global_load_async_to_lds_b32 %0, %1, off"
               :: "v"(lds_off), "v"(ga) : "memory");
}
__device__ __forceinline__ void wait_asynccnt_le1() {
  asm volatile("s_wait_asynccnt 0x1" ::: "memory");
}
__device__ __forceinline__ void wait_asynccnt_le0() {
  asm volatile("s_wait_asynccnt 0x0" ::: "memory");
}

// =====================================================================
// Kernel 2: persistent recurrence. One workgroup per batch element;
// the 128x512 tape lives in LDS (padded stride 513) for all 256 steps.
// x_pre rows double-buffered in LDS via async global->LDS loads.
// =====================================================================
__global__ __launch_bounds__(512, 1) void e25_recurrent(
    const float* __restrict__ h_tape0,   // [B, NSLOT, DIM]
    const float* __restrict__ h_work0,   // [B, DIM]
    const float* __restrict__ W_h,       // [DIM, DIM]
    const float* __restrict__ W_write,   // [DIM, DIM]
    float* __restrict__ out)             // concat(h_all[B,T,D] (pre-filled x@Wx^T+b), tape, hwork)
{
  extern __shared__ float smem[];
  float* tape  = smem;                   // NSLOT * TS
  float* hwork = tape  + NSLOT * TS;     // DIM
  float* hnew  = hwork + DIM;            // DIM
  float* wv    = hnew  + DIM;            // DIM
  float* red   = wv    + DIM;            // DIM (partial-sum scratch)
  float* xbuf  = red   + DIM;            // 2 * DIM (async x_pre double buffer)
  float* z     = xbuf  + 2 * DIM;        // NSLOT
  float* zs    = z     + NSLOT;          // NSLOT
  float* tau   = zs    + NSLOT;          // NSLOT
  float* c1    = tau   + NSLOT;          // NSLOT
  float* c2    = c1    + NSLOT;          // NSLOT
  float* attn  = c2    + NSLOT;          // NSLOT
  float* scal  = attn  + NSLOT;          // 4
  int*   flg   = (int*)(scal + 4);       // NSLOT

  const int b   = blockIdx.x;
  const int tid = threadIdx.x;           // 0..511

  const size_t O1 = (size_t)BATCH * TSTEPS * DIM;       // start of h_tape out
  const size_t O2 = O1 + (size_t)BATCH * NSLOT * DIM;   // start of h_work out

  // ---- load initial state into LDS ----
  for (int i = tid; i < NSLOT * DIM; i += 512) {
    const int n = i >> 9, d = i & (DIM - 1);
    tape[n * TS + d] = h_tape0[(size_t)b * NSLOT * DIM + i];
  }
  hwork[tid] = h_work0[(size_t)b * DIM + tid];

  // ---- kick off async load of x_pre row for t=0 ----
  const float* xrow0 = out + (size_t)b * TSTEPS * DIM + tid;
  async_load_f32_to_lds(xrow0, &xbuf[tid]);
  __syncthreads();

  const int  sn = tid & (NSLOT - 1);     // slot for score partials
  const int  sq = tid >> 7;              // quarter of D for score partials
  const float* wh_row = W_h     + (size_t)tid * DIM;
  const float* ww_row = W_write + (size_t)tid * DIM;

  for (int t = 0; t < TSTEPS; ++t) {
    const int cur = t & 1, nxt = cur ^ 1;
    const bool has_next = (t + 1 < TSTEPS);
    if (has_next)   // prefetch next step's x_pre row one step ahead
      async_load_f32_to_lds(xrow0 + (size_t)(t + 1) * DIM, &xbuf[nxt * DIM + tid]);

    __builtin_prefetch(wh_row, 0, 3);
    __builtin_prefetch(ww_row, 0, 3);

    // ---------- read scores: z[n] = (tape[n,:] . hwork) * scale/2 ----------
    {
      const float* tr = tape  + sn * TS + sq * 128;
      const float* hv = hwork + sq * 128;
      float acc = 0.f;
#pragma unroll 8
      for (int i = 0; i < 128; ++i) acc = fmaf(tr[i], hv[i], acc);
      red[tid] = acc;
    }
    __syncthreads();
    if (tid < NSLOT)
      z[tid] = (red[tid] + red[tid + 128] + red[tid + 256] + red[tid + 384]) * ZSCALE;
    block_entmax15(tid, z, zs, tau, c1, c2, flg, attn, scal);

    // ---------- read vector (sparse over support) ----------
    float readv = 0.f;
#pragma unroll 4
    for (int n = 0; n < NSLOT; ++n) {
      const float a = attn[n];
      if (a != 0.f) readv = fmaf(a, tape[n * TS + tid], readv);
    }

    // ---------- pre = hwork @ W_h^T + x_pre(+bias) + read; h_new = tanh ----------
    float acc = 0.f;
#pragma unroll 4
    for (int d = 0; d < DIM; d += 4) {
      const float4 w = *(const float4*)(wh_row + d);
      acc = fmaf(hwork[d + 0], w.x, acc);
      acc = fmaf(hwork[d + 1], w.y, acc);
      acc = fmaf(hwork[d + 2], w.z, acc);
      acc = fmaf(hwork[d + 3], w.w, acc);
    }
    // wait for this step's async x_pre row (next-step prefetch may stay in flight;
    // async loads complete in order, so asynccnt<=1 retires the current one)
    if (has_next) wait_asynccnt_le1(); else wait_asynccnt_le0();
    const float xp = xbuf[cur * DIM + tid];           // own slot, own async load
    const float hn = tanhf(acc + xp + readv);
    __syncthreads();
    hnew[tid] = hn;
    out[(size_t)b * TSTEPS * DIM + (size_t)t * DIM + tid] = hn;  // h_work_all[b,t,:]
    __syncthreads();

    // ---------- write value: wv = h_new @ W_write^T ----------
    float accw = 0.f;
#pragma unroll 4
    for (int d = 0; d < DIM; d += 4) {
      const float4 w = *(const float4*)(ww_row + d);
      accw = fmaf(hnew[d + 0], w.x, accw);
      accw = fmaf(hnew[d + 1], w.y, accw);
      accw = fmaf(hnew[d + 2], w.z, accw);
      accw = fmaf(hnew[d + 3], w.w, accw);
    }
    wv[tid] = accw;

    // ---------- write scores ----------
    {
      const float* tr = tape + sn * TS + sq * 128;
      const float* hv = hnew + sq * 128;
      float a2 = 0.f;
#pragma unroll 8
      for (int i = 0; i < 128; ++i) a2 = fmaf(tr[i], hv[i], a2);
      red[tid] = a2;
    }
    __syncthreads();
    if (tid < NSLOT)
      z[tid] = (red[tid] + red[tid + 128] + red[tid + 256] + red[tid + 384]) * ZSCALE;
    block_entmax15(tid, z, zs, tau, c1, c2, flg, attn, scal);

    // ---------- tape convex blend (skip zero-attention slots) ----------
    const float mywv = wv[tid];
#pragma unroll 4
    for (int n = 0; n < NSLOT; ++n) {
      const float a = attn[n];
      if (a != 0.f) {
        const float tv = tape[n * TS + tid];
        tape[n * TS + tid] = fmaf(a, mywv - tv, tv);
      }
    }
    hwork[tid] = hn;
    __syncthreads();
  }

  // ---- final state to global ----
  for (int i = tid; i < NSLOT * DIM; i += 512) {
    const int n = i >> 9, d = i & (DIM - 1);
    out[O1 + (size_t)b * NSLOT * DIM + i] = tape[n * TS + d];
  }
  out[O2 + (size_t)b * DIM + tid] = hwork[tid];
}

// =====================================================================
// host launcher
// =====================================================================
extern "C" void kernel_launch(void* const* d_in, const int* in_sizes, int n_in,
                              void* d_out, int out_size, void* d_ws, size_t ws_size,
                              hipStream_t stream) {
  (void)in_sizes; (void)n_in; (void)out_size; (void)d_ws; (void)ws_size;
  const float* x_seq   = (const float*)d_in[0];
  const float* h_tape0 = (const float*)d_in[1];
  const float* h_work0 = (const float*)d_in[2];
  const float* W_h     = (const float*)d_in[3];
  const float* W_x     = (const float*)d_in[4];
  const float* b_h     = (const float*)d_in[5];
  const float* W_write = (const float*)d_in[6];
  float* out = (float*)d_out;

  // Kernel 1: x_pre = x_seq @ W_x^T + b_h -> h_work_all region of d_out.
  // 4096 row-tiles * 8 col-groups(of 4 tiles) = 32768 waves / 8 per block.
  const int n_blocks = (BATCH * TSTEPS / 16) * (DIM / 16 / NT) / 8;
  xw_gemm_wmma4<<<n_blocks, 256, 0, stream>>>(x_seq, W_x, b_h, out);

  // Kernel 2: persistent recurrence, one WG per batch element, tape in LDS.
  const size_t smem_floats = (size_t)NSLOT * TS   // tape
                           + 4 * DIM              // hwork, hnew, wv, red
                           + 2 * DIM              // xbuf double buffer
                           + 6 * NSLOT            // z, zs, tau, c1, c2, attn
                           + 4                    // scal
                           + NSLOT;               // flg (ints)
  const size_t smem_bytes = smem_floats * sizeof(float);
  (void)hipFuncSetAttribute((const void*)e25_recurrent,
                            hipFuncAttributeMaxDynamicSharedMemorySize,
                            (int)smem_bytes);
  e25_recurrent<<<BATCH, 512, smem_bytes, stream>>>(h_tape0, h_work0, W_h, W_write, out);
}